// HGNN_42614665511184
// MI455X (gfx1250) — compile-verified
//
#include <hip/hip_runtime.h>
#include <hip/hip_bf16.h>

typedef float v2f __attribute__((ext_vector_type(2)));
typedef float v8f __attribute__((ext_vector_type(8)));

#define HG_N_NODES 100000
#define HG_N_EDGES 20000
#define HG_NNZ     1600000

// ---------------------------------------------------------------------------
// Degree accumulation: d_V[node]++ , d_E[edge]++
// ---------------------------------------------------------------------------
__global__ __launch_bounds__(256) void hg_deg_kernel(
    const int* __restrict__ node_idx, const int* __restrict__ edge_idx,
    float* __restrict__ dV, float* __restrict__ dE, int nnz)
{
    int i = blockIdx.x * 256 + threadIdx.x;
    if (i < nnz) {
        atomicAdd(&dV[node_idx[i]], 1.0f);
        atomicAdd(&dE[edge_idx[i]], 1.0f);
    }
}

// dV -> dV^-1/2 (in place), dE -> 1/dE (in place)
__global__ __launch_bounds__(256) void hg_finalize_deg(
    float* __restrict__ dV, float* __restrict__ dE, int nV, int nE)
{
    int i = blockIdx.x * 256 + threadIdx.x;
    if (i < nV) dV[i] = rsqrtf(dV[i]);
    if (i < nE) dE[i] = 1.0f / dE[i];
}

// ---------------------------------------------------------------------------
// WMMA f32 GEMM: Y[M x 64] = op(A[M x K]) @ W[K x 64] + bias
// K / RELU_A compile-time; K-loop fully unrolled. One wave computes a 16-row
// strip (4 accumulators of 16x16 across N=64). W staged in LDS once per block
// in K-pair-interleaved layout: ldsP[p*64+n] = (W[2p][n], W[2p+1][n]) so each
// B fragment is a single aligned ds_load_b64 into a consecutive VGPR pair.
// ---------------------------------------------------------------------------
template <int K, int RELU_A>
__global__ __launch_bounds__(256) void hg_gemm_bias_wmma(
    const float* __restrict__ A, const float* __restrict__ W,
    const float* __restrict__ bias, float* __restrict__ Y, int M)
{
    __shared__ v2f ldsP[(K / 2) * 64];

    const int tid = threadIdx.x;
    // cooperative staging of W[K x 64] into LDS, K-pair interleaved
#pragma unroll
    for (int it = 0; it < K * 64 / 256; ++it) {
        int idx = it * 256 + tid;
        int kk  = idx >> 6;        // row in W
        int n   = idx & 63;        // column
        ldsP[(kk >> 1) * 64 + n][kk & 1] = W[idx];
    }
    __syncthreads();

    const int lane  = tid & 31;
    const int wave  = tid >> 5;
    const int strip = blockIdx.x * 8 + wave;          // 16-row strip index
    const int nStrips = M >> 4;
    if (strip >= nStrips) return;                     // wave-uniform: EXEC all-1s below

    const int m0    = strip << 4;
    const int l15   = lane & 15;
    const int khalf = (lane >> 4) << 1;               // 0 for lanes 0-15, 2 for 16-31

    // init accumulators from bias (bias depends only on column)
    v8f acc[4];
#pragma unroll
    for (int t = 0; t < 4; ++t) {
        float bv = bias[t * 16 + l15];
#pragma unroll
        for (int r = 0; r < 8; ++r) acc[t][r] = bv;
    }

    const float* __restrict__ arow = A + (size_t)(m0 + l15) * K + khalf;
#pragma unroll
    for (int k = 0; k < K; k += 4) {
        // A fragment 16x4: lanes 0-15 hold K=k,k+1 ; lanes 16-31 hold K=k+2,k+3
        v2f a = *(const v2f*)(arow + k);
        if (RELU_A) { a.x = fmaxf(a.x, 0.0f); a.y = fmaxf(a.y, 0.0f); }

        const int pp = (k + khalf) >> 1;              // K-pair index for this lane half
#pragma unroll
        for (int t = 0; t < 4; ++t) {
            // B fragment 4x16: one aligned b64 load (mirrors A striping over K)
            v2f b = ldsP[pp * 64 + t * 16 + l15];
            acc[t] = __builtin_amdgcn_wmma_f32_16x16x4_f32(
                /*neg_a=*/false, a, /*neg_b=*/false, b,
                /*c_mod=*/(short)0, acc[t],
                /*reuse_a=*/false, /*reuse_b=*/false);
        }
    }

    // store: VGPR r -> row m0+r (lanes 0-15) / m0+r+8 (lanes 16-31), col t*16+l15
    const int rowAdd = (lane >> 4) * 8;
#pragma unroll
    for (int t = 0; t < 4; ++t) {
        float* yb = Y + (size_t)(m0 + rowAdd) * 64 + t * 16 + l15;
#pragma unroll
        for (int r = 0; r < 8; ++r) yb[(size_t)r * 64] = acc[t][r];
    }
}

// ---------------------------------------------------------------------------
// Scatter pass 1: Ee[e,:] += Y[n,:] * dv_invsqrt[n]   (64 cols per nnz pair)
// blockDim=256 -> 4 nnz entries per block, one lane per column.
// ---------------------------------------------------------------------------
__global__ __launch_bounds__(256) void hg_scatter_nodes_to_edges(
    const float* __restrict__ Y, const float* __restrict__ dvs,
    const int* __restrict__ node_idx, const int* __restrict__ edge_idx,
    float* __restrict__ Ee, int nnz)
{
    int i = blockIdx.x * 4 + (threadIdx.x >> 6);
    int c = threadIdx.x & 63;
    if (i >= nnz) return;
    int n = node_idx[i];
    int e = edge_idx[i];
    float v = Y[(size_t)n * 64 + c] * dvs[n];
    atomicAdd(&Ee[(size_t)e * 64 + c], v);
}

// ---------------------------------------------------------------------------
// Scatter pass 2: Z[n,:] += Ee[e,:] * de_inv[e] * dv_invsqrt[n]
// (final Dv^-1/2 scaling baked into each contribution)
// ---------------------------------------------------------------------------
__global__ __launch_bounds__(256) void hg_scatter_edges_to_nodes(
    const float* __restrict__ Ee, const float* __restrict__ dei,
    const float* __restrict__ dvs,
    const int* __restrict__ node_idx, const int* __restrict__ edge_idx,
    float* __restrict__ Z, int nnz)
{
    int i = blockIdx.x * 4 + (threadIdx.x >> 6);
    int c = threadIdx.x & 63;
    if (i >= nnz) return;
    int n = node_idx[i];
    int e = edge_idx[i];
    float v = Ee[(size_t)e * 64 + c] * dei[e] * dvs[n];
    atomicAdd(&Z[(size_t)n * 64 + c], v);
}

// ---------------------------------------------------------------------------
// Host-side orchestration
// ---------------------------------------------------------------------------
extern "C" void kernel_launch(void* const* d_in, const int* in_sizes, int n_in,
                              void* d_out, int out_size, void* d_ws, size_t ws_size,
                              hipStream_t stream) {
    const float* X        = (const float*)d_in[0];
    const float* W1       = (const float*)d_in[1];
    const float* b1       = (const float*)d_in[2];
    const float* W2       = (const float*)d_in[3];
    const float* b2       = (const float*)d_in[4];
    const int*   node_idx = (const int*)d_in[5];
    const int*   edge_idx = (const int*)d_in[6];
    float* out = (float*)d_out;

    // workspace layout (floats): Y | Z | Ee | dV | dE   (~57 MB total)
    float* ws = (float*)d_ws;
    float* Y  = ws;
    float* Z  = Y  + (size_t)HG_N_NODES * 64;
    float* Ee = Z  + (size_t)HG_N_NODES * 64;
    float* dV = Ee + (size_t)HG_N_EDGES * 64;
    float* dE = dV + HG_N_NODES;

    // degrees
    hipMemsetAsync(dV, 0, HG_N_NODES * sizeof(float), stream);
    hipMemsetAsync(dE, 0, HG_N_EDGES * sizeof(float), stream);
    hg_deg_kernel<<<HG_NNZ / 256, 256, 0, stream>>>(node_idx, edge_idx, dV, dE, HG_NNZ);
    hg_finalize_deg<<<(HG_N_NODES + 255) / 256, 256, 0, stream>>>(dV, dE, HG_N_NODES, HG_N_EDGES);

    const int nStrips = HG_N_NODES / 16;           // 6250
    const int gemmGrid = (nStrips + 7) / 8;        // 8 waves per block

    // layer 1 linear: Y = X @ W1 + b1
    hg_gemm_bias_wmma<128, 0><<<gemmGrid, 256, 0, stream>>>(X, W1, b1, Y, HG_N_NODES);

    // L_apply #1 -> Z  (Z = Dv^-1/2 H De^-1 H^T Dv^-1/2 Y)
    hipMemsetAsync(Ee, 0, (size_t)HG_N_EDGES * 64 * sizeof(float), stream);
    hg_scatter_nodes_to_edges<<<HG_NNZ / 4, 256, 0, stream>>>(Y, dV, node_idx, edge_idx, Ee, HG_NNZ);
    hipMemsetAsync(Z, 0, (size_t)HG_N_NODES * 64 * sizeof(float), stream);
    hg_scatter_edges_to_nodes<<<HG_NNZ / 4, 256, 0, stream>>>(Ee, dE, dV, node_idx, edge_idx, Z, HG_NNZ);

    // layer 2 linear with fused ReLU on input: Y = relu(Z) @ W2 + b2
    hg_gemm_bias_wmma<64, 1><<<gemmGrid, 256, 0, stream>>>(Z, W2, b2, Y, HG_N_NODES);

    // L_apply #2 -> d_out (final Dv^-1/2 scaling fused into scatter)
    hipMemsetAsync(Ee, 0, (size_t)HG_N_EDGES * 64 * sizeof(float), stream);
    hg_scatter_nodes_to_edges<<<HG_NNZ / 4, 256, 0, stream>>>(Y, dV, node_idx, edge_idx, Ee, HG_NNZ);
    hipMemsetAsync(out, 0, (size_t)HG_N_NODES * 64 * sizeof(float), stream);
    hg_scatter_edges_to_nodes<<<HG_NNZ / 4, 256, 0, stream>>>(Ee, dE, dV, node_idx, edge_idx, out, HG_NNZ);
}